// EffDet_46385646797679
// MI455X (gfx1250) — compile-verified
//
#include <hip/hip_runtime.h>
#include <stdint.h>

// ---------------- problem constants (from the reference) ----------------
#define NLVL 5
#define BATCH 16
#define NGT 32
#define A_TOTAL 49104            // 16-aligned; every level boundary is too
#define OFF1 36864
#define OFF2 46080
#define OFF3 48384
#define OFF4 48960

#define BLK 256                  // 8 wave32 waves per workgroup

typedef __attribute__((ext_vector_type(4))) uint32_t u32x4;
typedef __attribute__((ext_vector_type(8))) uint32_t u32x8;
typedef __attribute__((ext_vector_type(2))) float    v2f;
typedef __attribute__((ext_vector_type(8))) float    v8f;

static __device__ __forceinline__ int level_of(int a, int& off) {
  if (a < OFF1) { off = 0;    return 0; }
  if (a < OFF2) { off = OFF1; return 1; }
  if (a < OFF3) { off = OFF2; return 2; }
  if (a < OFF4) { off = OFF3; return 3; }
  off = OFF4; return 4;
}

// ---- Tensor Data Mover: stage n contiguous f32 elements global -> LDS ----
// Field-accurate D# (group0 128b + group1 256b, 1D tile) per
// CDNA5 ISA; inline asm form is portable across the
// ROCm7.2 / amdgpu-toolchain builtin-arity skew. One wave issues, then
// __syncthreads() publishes to the workgroup.
static __device__ __forceinline__ void tdm_stage_f32(uint32_t lds_off,
                                                     uint64_t gaddr,
                                                     uint32_t n) {
  u32x4 g0;
  u32x8 g1;
  g0[0] = 1u;                                   // count=1 (valid), user mode
  g0[1] = lds_off;                              // lds_addr (bytes)
  g0[2] = (uint32_t)gaddr;                      // global_addr[31:0]
  g0[3] = (uint32_t)((gaddr >> 32) & 0x01FFFFFFu) | (2u << 30); // [56:32], type=2
  g1[0] = (2u << 16);                           // data_size=4B
  g1[1] = (n & 0xFFFFu) << 16;                  // tensor_dim0[15:0]
  g1[2] = ((n >> 16) & 0xFFFFu) | (1u << 16);   // tensor_dim0[31:16], dim1=1
  g1[3] = (n & 0xFFFFu) << 16;                  // tile_dim0 = n
  g1[4] = 1u;                                   // tile_dim1=1
  g1[5] = n;                                    // tensor_dim0_stride
  g1[6] = 0u;
  g1[7] = 0u;
  asm volatile("tensor_load_to_lds %0, %1" :: "s"(g0), "s"(g1) : "memory");
  __builtin_amdgcn_s_wait_tensorcnt(0);
}

// ---- WMMA outer-sum: D[m][n] = a_m + b_n over a 16x16 tile ---------------
// A (16x4, documented layout): VGPR0 lanes0-15 = K0 = a_m; VGPR1 lanes16-31
// = K3 = 1; all other K slots zero. B uses only K0 (=1s, VGPR0 lanes0-15)
// and K3 (=b_n, VGPR1 lanes16-31) -- the two K slots whose VGPR placement is
// identical under either plausible 4x16 B striping, so the result is
// layout-unambiguous: D[m][n] = a_m*1 + 1*b_n.
static __device__ __forceinline__ v8f wmma_outer(float a_m, float b_n,
                                                 bool lolane) {
  v2f A, Bv;
  v8f C = {};
  A[0]  = lolane ? a_m  : 0.0f;   // K0 (low) | K2 (high)=0
  A[1]  = lolane ? 0.0f : 1.0f;   // K1=0    | K3=1
  Bv[0] = lolane ? 1.0f : 0.0f;   // K0 row = ones
  Bv[1] = lolane ? 0.0f : b_n;    // K3 row = b_n
  return __builtin_amdgcn_wmma_f32_16x16x4_f32(false, A, false, Bv,
                                               (short)0, C, false, false);
}

// IoU of anchor vs box (exact scalar form, used for output-determining pass)
static __device__ __forceinline__ float iou_f(const float4& a, const float4& b,
                                              float areaA) {
  float iw = fminf(a.z, b.z) - fmaxf(a.x, b.x);
  float ih = fminf(a.w, b.w) - fmaxf(a.y, b.y);
  iw = fmaxf(iw, 0.0f);
  ih = fmaxf(ih, 0.0f);
  float inter = iw * ih;
  float uni = areaA + (b.z - b.x) * (b.w - b.y) - inter;
  return inter / uni;
}

// ---------------- kernel 0: zero the packed argmax table -----------------
__global__ void k_init(unsigned long long* __restrict__ ws_pk, int n) {
  int t = blockIdx.x * blockDim.x + threadIdx.x;
  if (t < n) ws_pk[t] = 0ull;
}

// ------- kernel 1: WMMA-tiled per-(b,level,g) packed max over anchors ----
// packed = iou_bits<<32 | (0xFFFFFFFF - local_anchor_idx): IoU>=0 so float
// bits are order-preserving; ties resolve to LOWEST anchor index, matching
// jnp.argmax. Each wave: 32 anchors x 32 GTs as four 16x16 WMMA tile-pairs;
// column (=GT) reduction falls out of the C/D layout (column n lives in
// lanes n and n+16; row index = static VGPR index).
__global__ void k_pass1(const float* __restrict__ anchors,
                        const float* __restrict__ bb,
                        unsigned long long* __restrict__ ws_pk) {
  __shared__ float4 sbox[NGT];
  __shared__ unsigned long long spk[NLVL * NGT];
  const int b    = blockIdx.y;
  const int lane = threadIdx.x & 31;
  const int wv   = threadIdx.x >> 5;
  const bool lo  = lane < 16;

  if (threadIdx.x < 32) {  // wave 0 drives the TDM tile load (32*4 f32)
    tdm_stage_f32((uint32_t)(uintptr_t)&sbox[0],
                  (uint64_t)(uintptr_t)(bb + (size_t)b * NGT * 4), NGT * 4);
  }
  for (int t = threadIdx.x; t < NLVL * NGT; t += blockDim.x) spk[t] = 0ull;
  __syncthreads();

  // per-lane anchor (lane L of wave w -> anchor base+32w+L), clamped OOB
  const int wbase = blockIdx.x * BLK + wv * 32;
  int aidx = wbase + lane;
  if (aidx >= A_TOTAL) aidx = A_TOTAL - 1;
  const float4 an = ((const float4*)anchors)[aidx];
  const float aw = an.z - an.x, ah = an.w - an.y;
  const float areaA = aw * ah;

  // per-lane GT (lane L -> gt L)
  const float4 gb = sbox[lane];
  const float bw = gb.z - gb.x, bh = gb.w - gb.y;
  const float areaB = bw * bh;

  // half-swapped copies for tile selection (rt=1 rows / ct=0 cols)
  const float s_aw = __shfl_xor(aw, 16),   s_ah = __shfl_xor(ah, 16);
  const float s_x1 = __shfl_xor(an.x, 16), s_y1 = __shfl_xor(an.y, 16);
  const float s_x2 = __shfl_xor(an.z, 16), s_y2 = __shfl_xor(an.w, 16);
  const float s_aA = __shfl_xor(areaA, 16);
  const float t_bw = __shfl_xor(bw, 16),   t_bh = __shfl_xor(bh, 16);
  const float t_x1 = __shfl_xor(gb.x, 16), t_y1 = __shfl_xor(gb.y, 16);
  const float t_x2 = __shfl_xor(gb.z, 16), t_y2 = __shfl_xor(gb.w, 16);
  const float t_aB = __shfl_xor(areaB, 16);

#pragma unroll
  for (int rt = 0; rt < 2; ++rt) {
    const int tbase = wbase + rt * 16;            // 16-aligned => whole tile
    if (tbase >= A_TOTAL) break;                  // in one level or absent
    int off;
    const int li = level_of(tbase, off);
    const uint32_t lbase = (uint32_t)(tbase - off);

    const float Aaw = rt ? s_aw : aw,  Aah = rt ? s_ah : ah;
    const float Ax1 = rt ? s_x1 : an.x, Ay1 = rt ? s_y1 : an.y;
    const float Ax2 = rt ? s_x2 : an.z, Ay2 = rt ? s_y2 : an.w;
    const float AaA = rt ? s_aA : areaA;

#pragma unroll
    for (int ct = 0; ct < 2; ++ct) {
      const float Bbw = ct ? bw : t_bw,  Bbh = ct ? bh : t_bh;
      const float Bx1 = ct ? gb.x : t_x1, By1 = ct ? gb.y : t_y1;
      const float Bx2 = ct ? gb.z : t_x2, By2 = ct ? gb.w : t_y2;
      const float BaB = ct ? areaB : t_aB;

      // 7 outer-sum WMMAs: widths, +/- coordinate diffs, union base
      v8f Wx  = wmma_outer(Aaw, Bbw,  lo);
      v8f Dx1 = wmma_outer(Ax1, -Bx1, lo);
      v8f Dx2 = wmma_outer(Ax2, -Bx2, lo);
      v8f Wy  = wmma_outer(Aah, Bbh,  lo);
      v8f Dy1 = wmma_outer(Ay1, -By1, lo);
      v8f Dy2 = wmma_outer(Ay2, -By2, lo);
      v8f U0  = wmma_outer(AaA, BaB,  lo);

      unsigned long long acc = 0ull;
#pragma unroll
      for (int r = 0; r < 8; ++r) {
        // iw = min(ax2,bx2)-max(ax1,bx1) = 0.5*(Wx - |Dx1| - |Dx2|)
        float iw = 0.5f * (Wx[r] - fabsf(Dx1[r]) - fabsf(Dx2[r]));
        float ih = 0.5f * (Wy[r] - fabsf(Dy1[r]) - fabsf(Dy2[r]));
        iw = fmaxf(iw, 0.0f);
        ih = fmaxf(ih, 0.0f);
        float inter = iw * ih;
        float iou = inter / (U0[r] - inter);
        uint32_t row = lbase + (uint32_t)r + (lo ? 0u : 8u);
        unsigned long long pk =
            ((unsigned long long)__float_as_uint(iou) << 32) |
            (unsigned long long)(0xFFFFFFFFu - row);
        acc = (pk > acc) ? pk : acc;
      }
      // column n lives in lanes n and n+16: one xor-16 combine
      unsigned long long other = __shfl_xor(acc, 16);
      acc = (other > acc) ? other : acc;
      if (lo) atomicMax(&spk[li * NGT + ct * 16 + lane], acc);  // ds_max_u64
    }
  }
  __syncthreads();

  for (int t = threadIdx.x; t < NLVL * NGT; t += blockDim.x) {
    unsigned long long v = spk[t];
    if (v) atomicMax(&ws_pk[(size_t)b * NLVL * NGT + t], v);  // global u64 max
  }
}

// ---------------- kernel 2: per-GT best level (first max wins) -----------
__global__ void k_levels(const unsigned long long* __restrict__ ws_pk,
                         uint32_t* __restrict__ ws_lvl) {
  int t = blockIdx.x * blockDim.x + threadIdx.x;
  if (t >= BATCH * NGT) return;
  const int b = t / NGT, g = t % NGT;
  uint32_t best = (uint32_t)(ws_pk[(size_t)b * NLVL * NGT + g] >> 32);
  int lvl = 0;
  for (int li = 1; li < NLVL; ++li) {
    uint32_t v = (uint32_t)(ws_pk[(size_t)b * NLVL * NGT + li * NGT + g] >> 32);
    if (v > best) { best = v; lvl = li; }
  }
  ws_lvl[t] = (uint32_t)lvl;
}

// ---------------- kernel 3: final assignment (exact scalar) --------------
__global__ void k_assign(const float* __restrict__ anchors,
                         const float* __restrict__ bb,
                         const int* __restrict__ ids,
                         const unsigned long long* __restrict__ ws_pk,
                         const uint32_t* __restrict__ ws_lvl,
                         float* __restrict__ out_labels,
                         float* __restrict__ out_reg) {
  __shared__ float4 sbox[NGT];
  __shared__ int sid[NGT];
  __shared__ uint32_t slvl[NGT];
  __shared__ uint32_t sga[NLVL * NGT];
  const int b = blockIdx.y;
  const int a = blockIdx.x * blockDim.x + threadIdx.x;

  if (threadIdx.x < 32) {
    tdm_stage_f32((uint32_t)(uintptr_t)&sbox[0],
                  (uint64_t)(uintptr_t)(bb + (size_t)b * NGT * 4), NGT * 4);
  }
  for (int t = threadIdx.x; t < NGT; t += blockDim.x) {
    sid[t]  = ids[b * NGT + t];
    slvl[t] = ws_lvl[b * NGT + t];
  }
  for (int t = threadIdx.x; t < NLVL * NGT; t += blockDim.x) {
    sga[t] = 0xFFFFFFFFu -
             (uint32_t)(ws_pk[(size_t)b * NLVL * NGT + t] & 0xFFFFFFFFull);
  }
  __syncthreads();
  if (a >= A_TOTAL) return;

  int off;
  const int li = level_of(a, off);
  const uint32_t al = (uint32_t)(a - off);
  const float4 an = ((const float4*)anchors)[a];
  const float aw = an.z - an.x, ah = an.w - an.y;
  const float areaA = aw * ah;

  // per-anchor max IoU / argmax GT (first max wins)
  float mo = -1.0f;
  int am = 0;
  for (int g = 0; g < NGT; ++g) {
    float v = iou_f(an, sbox[g], areaA);
    if (v > mo) { mo = v; am = g; }
  }

  // .at[ga].set applies in g order (last write wins; non-is_max duplicate
  // writes restore boxes[am]); pos uses scatter-max (any is_max).
  bool anymax = false;
  int ovr = -1;
  for (int g = 0; g < NGT; ++g) {
    if (sga[li * NGT + g] == al) {
      bool im = (slvl[g] == (uint32_t)li);
      anymax |= im;
      ovr = im ? g : -1;
    }
  }

  const bool pos = (mo >= 0.5f) || anymax;
  const int src = (ovr >= 0) ? ovr : am;

  float lab;
  if (pos)             lab = (float)sid[src];
  else if (mo >= 0.4f) lab = -1.0f;   // mo < 0.5 here since pos is false
  else                 lab = 0.0f;

  const size_t oi = (size_t)b * A_TOTAL + a;
  out_labels[oi] = lab;

  float4 r = make_float4(0.0f, 0.0f, 0.0f, 0.0f);
  if (pos) {
    const float4 bx = sbox[src];
    float acx = an.x + 0.5f * aw, acy = an.y + 0.5f * ah;
    float gw = bx.z - bx.x, gh = bx.w - bx.y;
    float gcx = bx.x + 0.5f * gw, gcy = bx.y + 0.5f * gh;
    r.x = (gcx - acx) / aw;
    r.y = (gcy - acy) / ah;
    r.z = logf(gw / aw);
    r.w = logf(gh / ah);
  }
  ((float4*)out_reg)[oi] = r;
}

extern "C" void kernel_launch(void* const* d_in, const int* in_sizes, int n_in,
                              void* d_out, int out_size, void* d_ws, size_t ws_size,
                              hipStream_t stream) {
  (void)in_sizes; (void)n_in; (void)out_size; (void)ws_size;
  const float* anchors = (const float*)d_in[0];   // [A_TOTAL,4] f32
  const float* bb      = (const float*)d_in[1];   // [B*G,4]     f32
  const int*   birdids = (const int*)d_in[2];     // [B*G]       i32

  unsigned long long* ws_pk = (unsigned long long*)d_ws;            // B*5*32 u64
  uint32_t* ws_lvl = (uint32_t*)((char*)d_ws +
                                 (size_t)BATCH * NLVL * NGT * sizeof(unsigned long long));

  float* out_labels = (float*)d_out;                                // B*A
  float* out_reg    = out_labels + (size_t)BATCH * A_TOTAL;         // B*A*4

  const int npk = BATCH * NLVL * NGT;
  k_init<<<(npk + BLK - 1) / BLK, BLK, 0, stream>>>(ws_pk, npk);

  dim3 grid((A_TOTAL + BLK - 1) / BLK, BATCH);
  k_pass1<<<grid, BLK, 0, stream>>>(anchors, bb, ws_pk);
  k_levels<<<(BATCH * NGT + BLK - 1) / BLK, BLK, 0, stream>>>(ws_pk, ws_lvl);
  k_assign<<<grid, BLK, 0, stream>>>(anchors, bb, birdids, ws_pk, ws_lvl,
                                     out_labels, out_reg);
}